// ProcessContinuous_24704651887296
// MI455X (gfx1250) — compile-verified
//
#include <hip/hip_runtime.h>

typedef __attribute__((ext_vector_type(2))) float v2f;
typedef __attribute__((ext_vector_type(8))) float v8f;

#define E_DIM 512
#define NCOL (5 * E_DIM)                      // 2560 output columns
#define NTILES (NCOL / 16)                    // 160 n-tiles
#define WAVES_PER_MTILE 16
#define NT_PER_WAVE (NTILES / WAVES_PER_MTILE) // 10

// Fold biases + order_table into per-column tables in the interleaved j-layout:
//   j = e*5 + c ; coef[j] = w_c[e] ; bias[j] = b_c[e] + order_table[c][e]
__global__ __launch_bounds__(256) void pc_precompute(
    const float* __restrict__ w_bet, const float* __restrict__ b_bet,
    const float* __restrict__ w_stack, const float* __restrict__ b_stack,
    const float* __restrict__ w_call, const float* __restrict__ b_call,
    const float* __restrict__ w_odds, const float* __restrict__ b_odds,
    const float* __restrict__ order, float* __restrict__ coef,
    float* __restrict__ bias) {
  int j = blockIdx.x * blockDim.x + threadIdx.x;
  if (j >= NCOL) return;
  int e = j / 5;
  int c = j - 5 * e;
  float w, b;
  if (c == 0)      { w = w_bet[e];   b = b_bet[e];   }
  else if (c <= 2) { w = w_stack[e]; b = b_stack[e]; }   // stack_fc shared (hero/villain)
  else if (c == 3) { w = w_call[e];  b = b_call[e];  }
  else             { w = w_odds[e];  b = b_odds[e];  }
  coef[j] = w;
  bias[j] = b + order[c * E_DIM + e];
}

// out = x[16x5 (K padded to 8)] @ W[8 x 2560] + Bias, W column j has one nonzero
// coef[j] at row j%5.  Two chained v_wmma_f32_16x16x4_f32 per 16x16 tile.
__global__ __launch_bounds__(256) void pc_wmma_main(
    const float* __restrict__ x, const float* __restrict__ coef,
    const float* __restrict__ bias, float* __restrict__ out) {
  const int lane  = threadIdx.x & 31;
  const int wave  = threadIdx.x >> 5;                 // 0..7
  const int mTile = blockIdx.x >> 1;
  const int wgid  = ((blockIdx.x & 1) << 3) | wave;   // 0..15: wave slot within m-tile
  const int m0    = mTile << 4;
  const int lrow  = lane & 15;                        // M for A, N for B/C/D
  const int hi    = lane >> 4;                        // lane-half selects K pair / row+8

  // A fragments: loop-invariant (x row reused across all n-tiles).
  // a1 covers K=0..3, a2 covers K=4..7 (k=5,6,7 are zero padding).
  const float* xr = x + (size_t)(m0 + lrow) * 5;
  v2f a1, a2;
  a1.x = xr[2 * hi];
  a1.y = xr[2 * hi + 1];
  a2.x = hi ? 0.0f : xr[4];
  a2.y = 0.0f;

  for (int i = 0; i < NT_PER_WAVE; ++i) {
    const int nTile = wgid + WAVES_PER_MTILE * i;     // 0..159
    const int j = (nTile << 4) + lrow;                // output column
    const int c = j % 5;
    const float cf = coef[j];
    const float bs = bias[j];
    const int kb0 = hi << 1;
    // B fragment: W[k][j] = (k == j%5) ? coef[j] : 0
    v2f b1, b2;
    b1.x = (c == kb0)     ? cf : 0.0f;                // k = 2*hi
    b1.y = (c == kb0 + 1) ? cf : 0.0f;                // k = 2*hi+1
    b2.x = ((hi == 0) && (c == 4)) ? cf : 0.0f;       // k = 4 (k=6 half is zero)
    b2.y = 0.0f;                                      // k = 5 / 7
    // C fragment: bias[j] broadcast down the column (all 8 rows per lane).
    v8f acc;
#pragma unroll
    for (int r = 0; r < 8; ++r) acc[r] = bs;

    acc = __builtin_amdgcn_wmma_f32_16x16x4_f32(false, a2, false, b2,
                                                (short)0, acc, false, false);
    acc = __builtin_amdgcn_wmma_f32_16x16x4_f32(false, a1, false, b1,
                                                (short)0, acc, false, false);

    // D layout: VGPR r, lane -> row m0 + r + 8*hi, col j.
    float* o = out + (size_t)(m0 + (hi << 3)) * NCOL + j;
#pragma unroll
    for (int r = 0; r < 8; ++r)
      o[(size_t)r * NCOL] = acc[r];
  }
}

extern "C" void kernel_launch(void* const* d_in, const int* in_sizes, int n_in,
                              void* d_out, int out_size, void* d_ws, size_t ws_size,
                              hipStream_t stream) {
  const float* x       = (const float*)d_in[0];
  const float* w_bet   = (const float*)d_in[1];
  const float* b_bet   = (const float*)d_in[2];
  const float* w_stack = (const float*)d_in[3];
  const float* b_stack = (const float*)d_in[4];
  const float* w_call  = (const float*)d_in[5];
  const float* b_call  = (const float*)d_in[6];
  const float* w_odds  = (const float*)d_in[7];
  const float* b_odds  = (const float*)d_in[8];
  const float* order   = (const float*)d_in[9];
  float* out = (float*)d_out;

  float* coef = (float*)d_ws;        // 2560 floats
  float* bias = coef + NCOL;         // 2560 floats (20 KB total scratch)

  pc_precompute<<<(NCOL + 255) / 256, 256, 0, stream>>>(
      w_bet, b_bet, w_stack, b_stack, w_call, b_call, w_odds, b_odds,
      order, coef, bias);

  const int m = in_sizes[0] / 5;     // 65536 rows
  const int mTiles = m >> 4;         // 4096 m-tiles, 2 blocks (16 waves) each
  pc_wmma_main<<<mTiles * 2, 256, 0, stream>>>(x, coef, bias, out);
}